// GAT_43765716746408
// MI455X (gfx1250) — compile-verified
//
#include <hip/hip_runtime.h>
#include <math.h>

#define N_NODES 50000
#define N_EDGES 800000
#define D_MODEL 128
#define N_HEADS 2

typedef __attribute__((ext_vector_type(2))) float v2f;
typedef __attribute__((ext_vector_type(8))) float v8f;

// ---------- helpers ----------
__device__ __forceinline__ unsigned ordEnc(float f) {
    unsigned u = __float_as_uint(f);
    return (u & 0x80000000u) ? ~u : (u | 0x80000000u);
}
__device__ __forceinline__ float ordDec(unsigned u) {
    unsigned v = (u & 0x80000000u) ? (u & 0x7fffffffu) : ~u;
    return __uint_as_float(v);
}
#define ORD_NEG_INF 0x007FFFFFu   // ordEnc(-inf)

__global__ void fill_u32(unsigned* __restrict__ p, unsigned v, int n) {
    int i = blockIdx.x * blockDim.x + threadIdx.x;
    if (i < n) p[i] = v;
}

// ---------- GEMM: feat = h(50000x128) @ W(128x256), f32 WMMA ----------
// grid.x = N_NODES/16 (=3125), block = 128 (4 waves). Wave w computes C cols [w*64, w*64+64).
__global__ void __launch_bounds__(128) gemm_feat(const float* __restrict__ h,
                                                 const float* __restrict__ W,
                                                 float* __restrict__ feat) {
    const int lane  = threadIdx.x & 31;
    const int wave  = threadIdx.x >> 5;
    const int mbase = blockIdx.x << 4;
    const int nq    = wave << 6;          // 0,64,128,192
    const int row   = lane & 15;          // M (for A) / N (for B,C/D) within tile
    const int khalf = lane >> 4;          // which K pair this lane holds

    v8f acc0 = {}, acc1 = {}, acc2 = {}, acc3 = {};
    const float* Arow = h + (size_t)(mbase + row) * D_MODEL;

    for (int k = 0; k < D_MODEL; k += 4) {
        const int ka = k + khalf * 2;     // base K for this lane's pair
        v2f a;
        a.x = Arow[ka];
        a.y = Arow[ka + 1];
        const float* Wb = W + (size_t)ka * 256 + row;
        v2f b0, b1, b2, b3;
        b0.x = Wb[nq];        b0.y = Wb[256 + nq];
        b1.x = Wb[nq + 16];   b1.y = Wb[256 + nq + 16];
        b2.x = Wb[nq + 32];   b2.y = Wb[256 + nq + 32];
        b3.x = Wb[nq + 48];   b3.y = Wb[256 + nq + 48];
        acc0 = __builtin_amdgcn_wmma_f32_16x16x4_f32(false, a, false, b0, (short)0, acc0, false, false);
        acc1 = __builtin_amdgcn_wmma_f32_16x16x4_f32(false, a, false, b1, (short)0, acc1, false, false);
        acc2 = __builtin_amdgcn_wmma_f32_16x16x4_f32(false, a, false, b2, (short)0, acc2, false, false);
        acc3 = __builtin_amdgcn_wmma_f32_16x16x4_f32(false, a, false, b3, (short)0, acc3, false, false);
    }

    // C/D layout: VGPR j -> rows j (lanes 0-15) and j+8 (lanes 16-31), col = lane&15
    #pragma unroll
    for (int j = 0; j < 8; ++j) {
        const int m = mbase + j + khalf * 8;
        float* F = feat + (size_t)m * 256 + nq + row;
        F[0]  = acc0[j];
        F[16] = acc1[j];
        F[32] = acc2[j];
        F[48] = acc3[j];
    }
}

// ---------- per-node attention dots: el/er (one wave32 per node) ----------
__global__ void __launch_bounds__(256) node_attn(const float* __restrict__ feat,
                                                 const float* __restrict__ al,
                                                 const float* __restrict__ ar,
                                                 float* __restrict__ el,
                                                 float* __restrict__ er) {
    const int gw   = (blockIdx.x * blockDim.x + threadIdx.x) >> 5;
    const int lane = threadIdx.x & 31;
    if (gw >= N_NODES) return;
    const float4* F0 = (const float4*)(feat + (size_t)gw * 256) + lane;        // head 0
    const float4* F1 = (const float4*)(feat + (size_t)gw * 256 + 128) + lane;  // head 1
    const float4* AL0 = (const float4*)(al) + lane;
    const float4* AL1 = (const float4*)(al + 128) + lane;
    const float4* AR0 = (const float4*)(ar) + lane;
    const float4* AR1 = (const float4*)(ar + 128) + lane;
    float4 f0 = *F0, f1 = *F1;
    float4 l0 = *AL0, l1 = *AL1, r0 = *AR0, r1 = *AR1;
    float e0 = f0.x*l0.x + f0.y*l0.y + f0.z*l0.z + f0.w*l0.w;
    float e1 = f1.x*l1.x + f1.y*l1.y + f1.z*l1.z + f1.w*l1.w;
    float g0 = f0.x*r0.x + f0.y*r0.y + f0.z*r0.z + f0.w*r0.w;
    float g1 = f1.x*r1.x + f1.y*r1.y + f1.z*r1.z + f1.w*r1.w;
    #pragma unroll
    for (int off = 16; off; off >>= 1) {
        e0 += __shfl_xor(e0, off, 32);
        e1 += __shfl_xor(e1, off, 32);
        g0 += __shfl_xor(g0, off, 32);
        g1 += __shfl_xor(g1, off, 32);
    }
    if (lane == 0) {
        el[gw * 2 + 0] = e0;  el[gw * 2 + 1] = e1;
        er[gw * 2 + 0] = g0;  er[gw * 2 + 1] = g1;
    }
}

// ---------- edge pass 1: e = leaky_relu(el[src]+er[dst]); segment max ----------
__global__ void __launch_bounds__(256) edge_max(const float* __restrict__ el,
                                                const float* __restrict__ er,
                                                const int* __restrict__ src,
                                                const int* __restrict__ dst,
                                                float* __restrict__ ebuf,
                                                unsigned* __restrict__ nodemax) {
    int i = blockIdx.x * blockDim.x + threadIdx.x;
    if (i >= N_EDGES * N_HEADS) return;
    const int e = i >> 1, hh = i & 1;
    const int d = dst[e];
    float v = el[src[e] * 2 + hh] + er[d * 2 + hh];
    v = v > 0.0f ? v : 0.2f * v;
    ebuf[i] = v;
    atomicMax(nodemax + d * 2 + hh, ordEnc(v));
}

// ---------- edge pass 2: ex = exp(e - max[dst]); segment sum ----------
__global__ void __launch_bounds__(256) edge_exp(float* __restrict__ ebuf,
                                                const unsigned* __restrict__ nodemax,
                                                const int* __restrict__ dst,
                                                float* __restrict__ denom) {
    int i = blockIdx.x * blockDim.x + threadIdx.x;
    if (i >= N_EDGES * N_HEADS) return;
    const int e = i >> 1, hh = i & 1;
    const int d = dst[e];
    const float m = ordDec(nodemax[d * 2 + hh]);
    const float ex = expf(ebuf[i] - m);
    ebuf[i] = ex;
    unsafeAtomicAdd(denom + d * 2 + hh, ex);
}

// ---------- aggregation: agg[dst] += alpha * feat[src] (one wave per edge-head) ----------
__global__ void __launch_bounds__(256) aggregate(const float* __restrict__ feat,
                                                 const float* __restrict__ ebuf,
                                                 const float* __restrict__ denom,
                                                 const int* __restrict__ src,
                                                 const int* __restrict__ dst,
                                                 float* __restrict__ agg) {
    const int gw   = (blockIdx.x * blockDim.x + threadIdx.x) >> 5;
    const int lane = threadIdx.x & 31;
    if (gw >= N_EDGES * N_HEADS) return;
    const int e = gw >> 1, hh = gw & 1;
    const int s = src[e], d = dst[e];
    const float w = ebuf[gw] / denom[d * 2 + hh];
    const float4 f = *(const float4*)(feat + (size_t)s * 256 + hh * 128 + lane * 4);
    float* o = agg + (size_t)d * 256 + hh * 128 + lane * 4;
    unsafeAtomicAdd(o + 0, f.x * w);
    unsafeAtomicAdd(o + 1, f.y * w);
    unsafeAtomicAdd(o + 2, f.z * w);
    unsafeAtomicAdd(o + 3, f.w * w);
}

// ---------- finalize layer 1: hnext = mean over heads of (agg + b) ----------
__global__ void __launch_bounds__(256) finalize_mean(const float* __restrict__ agg,
                                                     const float* __restrict__ b,
                                                     float* __restrict__ hnext) {
    int i = blockIdx.x * blockDim.x + threadIdx.x;
    if (i >= N_NODES * D_MODEL) return;
    const int n = i >> 7, f = i & 127;
    const float o0 = agg[(size_t)n * 256 + f]       + b[f];
    const float o1 = agg[(size_t)n * 256 + 128 + f] + b[128 + f];
    hnext[i] = 0.5f * (o0 + o1);
}

// ---------- finalize layer 2: out += b (in place) ----------
__global__ void __launch_bounds__(256) finalize_bias(float* __restrict__ out,
                                                     const float* __restrict__ b) {
    int i = blockIdx.x * blockDim.x + threadIdx.x;
    if (i >= N_NODES * 256) return;
    out[i] += b[i & 255];
}

extern "C" void kernel_launch(void* const* d_in, const int* in_sizes, int n_in,
                              void* d_out, int out_size, void* d_ws, size_t ws_size,
                              hipStream_t stream) {
    const float* x   = (const float*)d_in[0];
    const float* Ws  = (const float*)d_in[1];  // (2,128,256)
    const float* als = (const float*)d_in[2];  // (2,2,128)
    const float* ars = (const float*)d_in[3];  // (2,2,128)
    const float* bs  = (const float*)d_in[4];  // (2,2,128)
    const int*   src = (const int*)d_in[5];
    const int*   dst = (const int*)d_in[6];

    // scratch layout (floats)
    float* ws      = (float*)d_ws;
    float* feat    = ws;                       // 12,800,000
    float* hnext   = ws + 12800000;            //  6,400,000
    float* el      = ws + 19200000;            //    100,000
    float* er      = ws + 19300000;            //    100,000
    unsigned* nmax = (unsigned*)(ws + 19400000); // 100,000
    float* denom   = ws + 19500000;            //    100,000
    float* ebuf    = ws + 19600000;            //  1,600,000
    // total 21,200,000 floats = 84.8 MB

    float* agg = (float*)d_out;                // 12,800,000 (used as agg buffer both layers)

    const int NH   = N_NODES * N_HEADS;        // 100,000
    const int EH   = N_EDGES * N_HEADS;        // 1,600,000
    const int NOUT = N_NODES * 256;            // 12,800,000

    for (int layer = 0; layer < 2; ++layer) {
        const float* hin = layer ? hnext : x;
        const float* W   = Ws  + (size_t)layer * D_MODEL * 256;
        const float* al  = als + (size_t)layer * 256;
        const float* ar  = ars + (size_t)layer * 256;
        const float* b   = bs  + (size_t)layer * 256;

        // init: agg=0, nodemax=ord(-inf), denom=0
        fill_u32<<<(NOUT + 255) / 256, 256, 0, stream>>>((unsigned*)agg, 0u, NOUT);
        fill_u32<<<(NH + 255) / 256, 256, 0, stream>>>(nmax, ORD_NEG_INF, NH);
        fill_u32<<<(NH + 255) / 256, 256, 0, stream>>>((unsigned*)denom, 0u, NH);

        gemm_feat<<<N_NODES / 16, 128, 0, stream>>>(hin, W, feat);
        node_attn<<<(N_NODES + 7) / 8, 256, 0, stream>>>(feat, al, ar, el, er);
        edge_max<<<(EH + 255) / 256, 256, 0, stream>>>(el, er, src, dst, ebuf, nmax);
        edge_exp<<<(EH + 255) / 256, 256, 0, stream>>>(ebuf, nmax, dst, denom);
        aggregate<<<(EH * 32 + 255) / 256, 256, 0, stream>>>(feat, ebuf, denom, src, dst, agg);

        if (layer == 0) {
            finalize_mean<<<(N_NODES * D_MODEL + 255) / 256, 256, 0, stream>>>(agg, b, hnext);
        } else {
            finalize_bias<<<(NOUT + 255) / 256, 256, 0, stream>>>(agg, b);
        }
    }
}